// GraphEMALayer_18133351924067
// MI455X (gfx1250) — compile-verified
//
#include <hip/hip_runtime.h>

// ---------------------------------------------------------------------------
// GraphEMALayer for MI455X (gfx1250, wave32).
//
// Roofline: dominant cost is streaming the per-edge message tensor
// M (E*64 fp32 = 409.6 MB) in and out of HBM for each of T=3 steps
// (~0.82 GB/step). Node-level arrays (h 25.6MB, m 2x25.6MB, deg 0.4MB ~78MB)
// must stay L2-resident (192 MB) so edge gathers and scatter-atomics hit
// on-chip. To protect that residency, the M stream / one-shot reads / final
// output use NON-TEMPORAL cache hints (CDNA5 TH=NT, ISA 7.3) so ~2.5 GB of
// streaming traffic does not evict the 78 MB random-access working set.
// Total HBM ~3.4 GB -> ~146 us at 23.3 TB/s. The GEMM (0.82 GFLOP) is
// negligible but uses V_WMMA_F32_16X16X4_F32 (fp32 WMMA): reference is fp32
// and the workload is bandwidth-bound, so no precision downconvert needed.
// ---------------------------------------------------------------------------

#define N_NODES 100000
#define N_EDGES 1600000
#define EH (N_EDGES / 2)
#define D 64
#define T_STEPS 3
#define EPSF 1e-9f

typedef float v2f __attribute__((ext_vector_type(2)));
typedef float v4f __attribute__((ext_vector_type(4)));
typedef float v8f __attribute__((ext_vector_type(8)));

// Non-temporal 128-bit load/store (emit global_load_b128/global_store_b128
// with TH=NT): bypass-ish L2 policy for streamed data.
__device__ __forceinline__ v4f load_nt(const float* p) {
  return __builtin_nontemporal_load(reinterpret_cast<const v4f*>(p));
}
__device__ __forceinline__ void store_nt(float* p, v4f v) {
  __builtin_nontemporal_store(v, reinterpret_cast<v4f*>(p));
}
// Regular-temporal 128-bit load for L2-resident arrays (h, m).
__device__ __forceinline__ v4f load_rt(const float* p) {
  return *reinterpret_cast<const v4f*>(p);
}

// ---------------------------------------------------------------------------
__global__ void fill_zero_kernel(float* __restrict__ p, size_t n) {
  size_t i = (size_t)blockIdx.x * blockDim.x + threadIdx.x;
  if (i < n) p[i] = 0.0f;
}

// ---------------------------------------------------------------------------
// h = x @ W + b  via V_WMMA_F32_16X16X4_F32.
// Block = 128 threads = 4 waves. Each block owns 16 rows of x; wave w owns
// output column tile [16w, 16w+16). x tile staged in LDS (padded stride 65
// to avoid 64-bank conflicts on the stride-64 A-fragment reads).
//
// Fragment layouts (ISA 7.12.2, 32-bit, wave32):
//   A 16x4 : lane l -> m = l&15, kHalf = 2*(l>>4); v0=A[m][k+kHalf], v1=A[m][k+kHalf+1]
//   B 4x16 : lane l -> n = l&15;                    v0=B[k+kHalf][n], v1=B[k+kHalf+1][n]
//   C 16x16: lane l -> n = l&15, row = vr + 8*(l>>4) for vr in 0..7
// ---------------------------------------------------------------------------
__global__ void gemm_bias_wmma_kernel(const float* __restrict__ x,
                                      const float* __restrict__ W,
                                      const float* __restrict__ bias,
                                      float* __restrict__ h) {
  __shared__ float xs[16 * 65];

  const int row0 = blockIdx.x * 16;
  const int tid  = threadIdx.x;

  // Stage 16x64 tile of x into LDS (1024 floats, 128 threads -> 8 each).
  #pragma unroll
  for (int i = 0; i < 8; ++i) {
    int idx = tid + i * 128;            // 0..1023
    int r = idx >> 6;                   // 0..15
    int c = idx & 63;                   // 0..63
    xs[r * 65 + c] = x[(size_t)(row0 + r) * D + c];
  }
  __syncthreads();

  const int wave  = tid >> 5;           // 0..3 -> column tile
  const int lane  = tid & 31;
  const int m     = lane & 15;
  const int n     = lane & 15;
  const int kHalf = (lane >> 4) * 2;    // 0 or 2
  const int col0  = wave * 16;

  v8f c = {};
  #pragma unroll
  for (int k0 = 0; k0 < D; k0 += 4) {
    const int ka = k0 + kHalf;
    v2f a, bfrag;
    a.x = xs[m * 65 + ka];
    a.y = xs[m * 65 + ka + 1];
    bfrag.x = W[(size_t)ka       * D + col0 + n];
    bfrag.y = W[(size_t)(ka + 1) * D + col0 + n];
    // (neg_a, A, neg_b, B, c_mod, C, reuse_a, reuse_b)
    c = __builtin_amdgcn_wmma_f32_16x16x4_f32(false, a, false, bfrag,
                                              (short)0, c, false, false);
  }

  const int   rowHalf = (lane >> 4) * 8;
  const float bv      = bias[col0 + n];
  #pragma unroll
  for (int vr = 0; vr < 8; ++vr) {
    const int r = row0 + rowHalf + vr;
    h[(size_t)r * D + col0 + n] = c[vr] + bv;
  }
}

// ---------------------------------------------------------------------------
// deg[n] = #edges with src==n (graph is symmetric, so this is the degree).
// ---------------------------------------------------------------------------
__global__ void degree_kernel(const int* __restrict__ src,
                              float* __restrict__ deg, int nE) {
  int e = blockIdx.x * blockDim.x + threadIdx.x;
  if (e < nE) atomicAdd(&deg[src[e]], 1.0f);
}

// ---------------------------------------------------------------------------
// M0[e] = h[src[e]] ; m0[dst[e]] += h[src[e]].  16 lanes per edge (float4).
// M0 store is non-temporal (write-once stream); h gather stays RT (L2 hot).
// ---------------------------------------------------------------------------
__global__ void init_msg_kernel(const float* __restrict__ h,
                                const int* __restrict__ src,
                                const int* __restrict__ dst,
                                float* __restrict__ M0,
                                float* __restrict__ m0, int nE) {
  const int t = blockIdx.x * blockDim.x + threadIdx.x;
  const int e = t >> 4;
  if (e >= nE) return;
  const int j4 = (t & 15) * 4;
  const int s = src[e];
  const int d = dst[e];

  const v4f hv = load_rt(h + (size_t)s * D + j4);
  store_nt(M0 + (size_t)e * D + j4, hv);

  float* mp = m0 + (size_t)d * D + j4;
  atomicAdd(mp + 0, hv.x);
  atomicAdd(mp + 1, hv.y);
  atomicAdd(mp + 2, hv.z);
  atomicAdd(mp + 3, hv.w);
}

// ---------------------------------------------------------------------------
// One EMA step:
//   M_new[e] = deg_e==1 ? h_e : (1-l)*h_e + l/(deg_e-1+eps) * (m[src[e]] - M[pair(e)])
//   m_new[dst[e]] += M_new[e]
// pair(e) = e+Eh for e<Eh else e-Eh  (the jnp.roll(-Eh)) -> contiguous stream.
// M traffic (read of pair, write of new) is non-temporal; h/m/deg gathers RT.
// ---------------------------------------------------------------------------
__global__ void ema_step_kernel(const float* __restrict__ h,
                                const float* __restrict__ deg,
                                const int* __restrict__ src,
                                const int* __restrict__ dst,
                                const float* __restrict__ Mcur,
                                const float* __restrict__ mcur,
                                float* __restrict__ Mnext,
                                float* __restrict__ mnext,
                                const float* __restrict__ lmbda_p, int nE) {
  const int t = blockIdx.x * blockDim.x + threadIdx.x;
  const int e = t >> 4;
  if (e >= nE) return;
  const int j4 = (t & 15) * 4;
  const int s = src[e];
  const int d = dst[e];
  const int ep = (e < EH) ? (e + EH) : (e - EH);

  const float lmbda = lmbda_p[0];
  const float dege  = deg[s];

  const v4f hv = load_rt(h + (size_t)s * D + j4);
  v4f out;
  if (dege == 1.0f) {
    out = hv;
  } else {
    const float coef = lmbda / (dege - 1.0f + EPSF);
    const float oml  = 1.0f - lmbda;
    const v4f mv  = load_rt(mcur + (size_t)s * D + j4);
    const v4f Mji = load_nt(Mcur + (size_t)ep * D + j4);
    out.x = oml * hv.x + coef * (mv.x - Mji.x);
    out.y = oml * hv.y + coef * (mv.y - Mji.y);
    out.z = oml * hv.z + coef * (mv.z - Mji.z);
    out.w = oml * hv.w + coef * (mv.w - Mji.w);
  }

  store_nt(Mnext + (size_t)e * D + j4, out);

  float* mp = mnext + (size_t)d * D + j4;
  atomicAdd(mp + 0, out.x);
  atomicAdd(mp + 1, out.y);
  atomicAdd(mp + 2, out.z);
  atomicAdd(mp + 3, out.w);
}

// ---------------------------------------------------------------------------
// out = x + relu( deg==0 ? h : (1-l)*h + l/(deg+eps)*m )
// x read-once and out write-once are non-temporal; h/m/deg are L2-hot (RT).
// ---------------------------------------------------------------------------
__global__ void finalize_kernel(const float* __restrict__ x,
                                const float* __restrict__ h,
                                const float* __restrict__ deg,
                                const float* __restrict__ m,
                                float* __restrict__ out,
                                const float* __restrict__ lmbda_p, int nN) {
  const int t = blockIdx.x * blockDim.x + threadIdx.x;
  const int node = t >> 4;
  if (node >= nN) return;
  const int j4 = (t & 15) * 4;

  const float lmbda = lmbda_p[0];
  const float dg    = deg[node];

  const v4f hv = load_rt(h + (size_t)node * D + j4);
  const v4f xv = load_nt(x + (size_t)node * D + j4);

  v4f y;
  if (dg == 0.0f) {
    y = hv;
  } else {
    const float oml  = 1.0f - lmbda;
    const float coef = lmbda / (dg + EPSF);
    const v4f mv = load_rt(m + (size_t)node * D + j4);
    y.x = oml * hv.x + coef * mv.x;
    y.y = oml * hv.y + coef * mv.y;
    y.z = oml * hv.z + coef * mv.z;
    y.w = oml * hv.w + coef * mv.w;
  }
  y.x = fmaxf(y.x, 0.0f);
  y.y = fmaxf(y.y, 0.0f);
  y.z = fmaxf(y.z, 0.0f);
  y.w = fmaxf(y.w, 0.0f);

  v4f o;
  o.x = xv.x + y.x;
  o.y = xv.y + y.y;
  o.z = xv.z + y.z;
  o.w = xv.w + y.w;
  store_nt(out + (size_t)node * D + j4, o);
}

// ---------------------------------------------------------------------------
extern "C" void kernel_launch(void* const* d_in, const int* in_sizes, int n_in,
                              void* d_out, int out_size, void* d_ws, size_t ws_size,
                              hipStream_t stream) {
  const float* x      = (const float*)d_in[0];
  const int*   eidx   = (const int*)  d_in[1];
  const float* W      = (const float*)d_in[2];
  const float* b      = (const float*)d_in[3];
  const float* lmbda  = (const float*)d_in[4];

  const int* src = eidx;              // edge_index[0] : first E entries
  const int* dst = eidx + N_EDGES;    // edge_index[1] : next E entries

  // Workspace layout (floats): h | deg | m0 | m1 | M0 | M1
  float* ws = (float*)d_ws;
  size_t off = 0;
  float* h   = ws + off; off += (size_t)N_NODES * D;
  float* deg = ws + off; off += (size_t)N_NODES;
  float* m0  = ws + off; off += (size_t)N_NODES * D;
  float* m1  = ws + off; off += (size_t)N_NODES * D;
  float* M0  = ws + off; off += (size_t)N_EDGES * D;
  float* M1  = ws + off; off += (size_t)N_EDGES * D;

  float* out = (float*)d_out;

  // Zero deg and m0 (accumulators).
  {
    size_t n = (size_t)N_NODES;
    fill_zero_kernel<<<(unsigned)((n + 255) / 256), 256, 0, stream>>>(deg, n);
  }
  {
    size_t n = (size_t)N_NODES * D;
    fill_zero_kernel<<<(unsigned)((n + 255) / 256), 256, 0, stream>>>(m0, n);
  }

  // h = x@W + b   (N_NODES is a multiple of 16: 6250 row tiles).
  gemm_bias_wmma_kernel<<<N_NODES / 16, 128, 0, stream>>>(x, W, b, h);

  // deg
  degree_kernel<<<(N_EDGES + 255) / 256, 256, 0, stream>>>(src, deg, N_EDGES);

  // M0 = h[src]; m0 = segment_sum(M0, dst)
  {
    size_t nth = (size_t)N_EDGES * 16;
    init_msg_kernel<<<(unsigned)((nth + 255) / 256), 256, 0, stream>>>(
        h, src, dst, M0, m0, N_EDGES);
  }

  // T EMA steps with double buffering.
  float* Mc = M0; float* Mn = M1;
  float* mc = m0; float* mn = m1;
  for (int t = 0; t < T_STEPS; ++t) {
    size_t nm = (size_t)N_NODES * D;
    fill_zero_kernel<<<(unsigned)((nm + 255) / 256), 256, 0, stream>>>(mn, nm);
    size_t nth = (size_t)N_EDGES * 16;
    ema_step_kernel<<<(unsigned)((nth + 255) / 256), 256, 0, stream>>>(
        h, deg, src, dst, Mc, mc, Mn, mn, lmbda, N_EDGES);
    float* tp;
    tp = Mc; Mc = Mn; Mn = tp;
    tp = mc; mc = mn; mn = tp;
  }

  // out = x + relu(EMA-combine(h, m))
  {
    size_t nth = (size_t)N_NODES * 16;
    finalize_kernel<<<(unsigned)((nth + 255) / 256), 256, 0, stream>>>(
        x, h, deg, mc, out, lmbda, N_NODES);
  }
}